// TimeSeriesNODE_5488968204429
// MI455X (gfx1250) — compile-verified
//
#include <hip/hip_runtime.h>

typedef float v2f __attribute__((ext_vector_type(2)));
typedef float v8f __attribute__((ext_vector_type(8)));

// Problem constants (from reference): x (B,N,T), W1 (D,N), W2 (D,D), W3 (N,D)
constexpr int kB = 16384;
constexpr int kN = 128;
constexpr int kT = 96;
constexpr int kH = 10;

constexpr int kWaves       = 4;              // waves per block (wave32)
constexpr int kThreads     = kWaves * 32;
constexpr int kRowsPerWave = 16;             // WMMA M
constexpr int kGrid        = kB / (kWaves * kRowsPerWave);   // 256 blocks

// LDS layout in float2 units.
// Weights transposed + pair-interleaved: Wp[k>>1][n] = (W[n][2kk], W[n][2kk+1])
constexpr int W1P_OFF = 0;                   // [64][64]  (K=128 pair rows x N=64)
constexpr int W2P_OFF = W1P_OFF + 64 * 64;   // [32][64]
constexpr int W3P_OFF = W2P_OFF + 32 * 64;   // [32][128]
constexpr int SCR_OFF = W3P_OFF + 32 * 128;
// per-wave: A-layout stage [64][16] + k1buf [32][32] + ybuf [32][32]
constexpr int STAGE_F2 = 64 * 16;            // 8KB
constexpr int KB_F2    = 32 * 32;            // 8KB each
constexpr int SCR_PER_WAVE = STAGE_F2 + 2 * KB_F2;
constexpr int SMEM_BYTES = (SCR_OFF + kWaves * SCR_PER_WAVE) * 8;  // 180224 B

struct St8 { v8f t[8]; };   // 16x128 tile in C/D layout: t[nt][v] = val[m0+v+8*hi][nt*16+l16]

__device__ __forceinline__ v8f wmma4(v2f a, v2f b, v8f c) {
  // D = A(16x4 f32) * B(4x16 f32) + C(16x16 f32)
  return __builtin_amdgcn_wmma_f32_16x16x4_f32(false, a, false, b, (short)0, c,
                                               false, false);
}

// Opaque zero: unknown-but-zero index bias. Added to LDS load indices so GVN
// cannot CSE / store-forward LDS loads across f-evals (which previously made the
// allocator cache LDS-resident values in scratch). Keeps pointers LDS-typed
// (unlike laundering the pointer, which degraded ds_load -> flat_load).
__device__ __forceinline__ int opaque_zero() {
  int z = 0;
  asm("" : "+v"(z));
  return z;
}

// tanh(x) = 1 - 2/(exp(2x)+1): ~5 VALU ops (v_exp_f32 + v_rcp_f32 transcendentals).
// Stable: x -> -inf: exp->0 -> -1 ; x -> +inf: exp->inf -> rcp->0 -> +1.
__device__ __forceinline__ float tanh_fast(float x) {
  const float e = __expf(2.0f * x);
  return 1.0f - 2.0f * __builtin_amdgcn_rcpf(e + 1.0f);
}

// Transpose-store one element of a D-layout tile into A-layout pair-interleaved
// stage buffer: element (k=cn, m=r) -> scrF[((cn>>1)*16 + r)*2 + (cn&1)]
__device__ __forceinline__ void tstore(float* scrF, int l16, int hi, int nt, int v,
                                       float val) {
  const int cn = nt * 16 + l16;
  const int r  = v + 8 * hi;
  scrF[(cn >> 1) * 32 + (cn & 1) + r * 2] = val;
}

// Bank-interleaved per-lane element buffers (ybuf / k1buf): pair (2j,2j+1) of tile nt.
// zs is an opaque zero (per staging loop) to defeat cross-f-eval CSE/forwarding.
__device__ __forceinline__ v2f ld2(const v2f* buf, int lane, int nt, int j, int zs) {
  return buf[(nt * 4 + j) * 32 + lane + zs];
}
__device__ __forceinline__ void st2(v2f* buf, int lane, int nt, int j, float a,
                                    float b) {
  v2f p;
  p.x = a;
  p.y = b;
  buf[(nt * 4 + j) * 32 + lane] = p;
}

// acc[nt] += A(16xK from stage) * B(KxNTO*16 from Wp)
// A frag: a[v] = A[l16][k0+v+2*hi]   -> scr[(k0/2+hi)*16 + l16]
// B frag: b[v] = B[k0+v+2*hi][n0+l16]-> Wp[(k0/2+hi)*NW + n0 + l16]
template <int K, int NTO, int NW>
__device__ __forceinline__ void mm(const v2f* scr, const v2f* Wp, int zo, int l16,
                                   int hi, v8f* acc) {
#pragma unroll
  for (int k0 = 0; k0 < K; k0 += 4) {
    const int kk = (k0 >> 1) + hi;
    const v2f a  = scr[kk * 16 + l16];
#pragma unroll
    for (int nt = 0; nt < NTO; ++nt) {
      const v2f b = Wp[kk * NW + nt * 16 + l16 + zo];
      acc[nt] = wmma4(a, b, acc[nt]);
    }
  }
}

// f(s) with s already staged in A-layout scratch.
// f(s) = tanh(tanh(s@W1.T+b1)@W2.T+b2)@W3.T+b3 for one 16-row tile.
__device__ __forceinline__ St8 feval_staged(const v2f* W1p, const v2f* W2p,
                                            const v2f* W3p, v2f* scr,
                                            const float* b1r, const float* b2r,
                                            const float* b3r, int l16, int hi) {
  float* scrF = (float*)scr;
  const int zo = opaque_zero();  // fresh index bias: no cross-f-eval weight CSE

  v8f h[4];
#pragma unroll
  for (int nt = 0; nt < 4; ++nt)
#pragma unroll
    for (int v = 0; v < 8; ++v) h[nt][v] = b1r[nt];
  mm<128, 4, 64>(scr, W1p, zo, l16, hi, h);  // 16x128 @ 128x64
#pragma unroll
  for (int nt = 0; nt < 4; ++nt)
#pragma unroll
    for (int v = 0; v < 8; ++v) h[nt][v] = tanh_fast(h[nt][v]);

  // h1 -> A layout (same-wave LDS ops are in-order: loads above precede stores)
#pragma unroll
  for (int nt = 0; nt < 4; ++nt)
#pragma unroll
    for (int v = 0; v < 8; ++v) tstore(scrF, l16, hi, nt, v, h[nt][v]);

  v8f h2[4];
#pragma unroll
  for (int nt = 0; nt < 4; ++nt)
#pragma unroll
    for (int v = 0; v < 8; ++v) h2[nt][v] = b2r[nt];
  mm<64, 4, 64>(scr, W2p, zo, l16, hi, h2);  // 16x64 @ 64x64
#pragma unroll
  for (int nt = 0; nt < 4; ++nt)
#pragma unroll
    for (int v = 0; v < 8; ++v) h2[nt][v] = tanh_fast(h2[nt][v]);

#pragma unroll
  for (int nt = 0; nt < 4; ++nt)
#pragma unroll
    for (int v = 0; v < 8; ++v) tstore(scrF, l16, hi, nt, v, h2[nt][v]);

  St8 o;
#pragma unroll
  for (int nt = 0; nt < 8; ++nt)
#pragma unroll
    for (int v = 0; v < 8; ++v) o.t[nt][v] = b3r[nt];
  mm<64, 8, 128>(scr, W3p, zo, l16, hi, o.t);  // 16x64 @ 64x128
  return o;
}

__global__ __launch_bounds__(kThreads, 1)
__attribute__((amdgpu_num_vgpr(384)))   // ~190 live + prefetch slack; no spills
void node_rk4_kernel(
    const float* __restrict__ x, const float* __restrict__ W1,
    const float* __restrict__ b1, const float* __restrict__ W2,
    const float* __restrict__ b2, const float* __restrict__ W3,
    const float* __restrict__ b3, float* __restrict__ out) {
  extern __shared__ v2f smem[];
  v2f* W1p = smem + W1P_OFF;
  v2f* W2p = smem + W2P_OFF;
  v2f* W3p = smem + W3P_OFF;

  const int tid  = threadIdx.x;
  const int wave = tid >> 5;
  const int lane = tid & 31;
  const int l16  = lane & 15;
  const int hi   = lane >> 4;

  // Cooperative, coalesced weight staging (global float2 index == loop index).
  const v2f* W1g = (const v2f*)W1;  // (64 x 128) -> rows of 64 float2
  for (int i = tid; i < 64 * 64; i += kThreads)
    W1p[(i & 63) * 64 + (i >> 6)] = W1g[i];
  const v2f* W2g = (const v2f*)W2;  // (64 x 64) -> rows of 32 float2
  for (int i = tid; i < 32 * 64; i += kThreads)
    W2p[(i & 31) * 64 + (i >> 5)] = W2g[i];
  const v2f* W3g = (const v2f*)W3;  // (128 x 64) -> rows of 32 float2
  for (int i = tid; i < 32 * 128; i += kThreads)
    W3p[(i & 31) * 128 + (i >> 5)] = W3g[i];

  float b1r[4], b2r[4], b3r[8];
#pragma unroll
  for (int nt = 0; nt < 4; ++nt) {
    b1r[nt] = b1[nt * 16 + l16];
    b2r[nt] = b2[nt * 16 + l16];
  }
#pragma unroll
  for (int nt = 0; nt < 8; ++nt) b3r[nt] = b3[nt * 16 + l16];

  __syncthreads();

  v2f* scr    = smem + SCR_OFF + wave * SCR_PER_WAVE;  // A-layout stage
  v2f* k1b    = scr + STAGE_F2;                        // k1 element buffer
  v2f* yb     = k1b + KB_F2;                           // y element buffer
  float* scrF = (float*)scr;
  const int m0 = (blockIdx.x * kWaves + wave) * kRowsPerWave;

  // y0 = x[:, :, T-1]: emit preds[...,0], park y in LDS, and stage s1 = y.
#pragma unroll
  for (int nt = 0; nt < 8; ++nt)
#pragma unroll
    for (int j = 0; j < 4; ++j) {
      float y0[2];
#pragma unroll
      for (int c = 0; c < 2; ++c) {
        const int v = 2 * j + c;
        y0[c] = x[(size_t)(m0 + v + 8 * hi) * (kN * kT) +
                  (size_t)(nt * 16 + l16) * kT + (kT - 1)];
        out[(size_t)(m0 + v + 8 * hi) * (kN * kH) + (size_t)(nt * 16 + l16) * kH] =
            y0[c];
        tstore(scrF, l16, hi, nt, v, y0[c]);
      }
      st2(yb, lane, nt, j, y0[0], y0[1]);
    }

  const float dt = 10.0f / 9.0f;
  for (int step = 1; step < kH; ++step) {
    // ---- k1 = f(y)  (s1 already staged by previous iteration / prologue)
    St8 K1 = feval_staged(W1p, W2p, W3p, scr, b1r, b2r, b3r, l16, hi);

    // ---- park k1 in LDS; stage s2 = y + (dt/3)*k1. K1 regs die here.
    {
      const int zs = opaque_zero();
#pragma unroll
      for (int nt = 0; nt < 8; ++nt)
#pragma unroll
        for (int j = 0; j < 4; ++j) {
          st2(k1b, lane, nt, j, K1.t[nt][2 * j], K1.t[nt][2 * j + 1]);
          const v2f yp = ld2(yb, lane, nt, j, zs);
#pragma unroll
          for (int c = 0; c < 2; ++c)
            tstore(scrF, l16, hi, nt, 2 * j + c,
                   yp[c] + (dt / 3.0f) * K1.t[nt][2 * j + c]);
        }
    }
    St8 K2 = feval_staged(W1p, W2p, W3p, scr, b1r, b2r, b3r, l16, hi);

    // ---- stage s3 = y + dt*k2 - (dt/3)*k1
    {
      const int zs = opaque_zero();
#pragma unroll
      for (int nt = 0; nt < 8; ++nt)
#pragma unroll
        for (int j = 0; j < 4; ++j) {
          const v2f yp  = ld2(yb, lane, nt, j, zs);
          const v2f k1p = ld2(k1b, lane, nt, j, zs);
#pragma unroll
          for (int c = 0; c < 2; ++c) {
            const int v = 2 * j + c;
            tstore(scrF, l16, hi, nt, v,
                   yp[c] + dt * K2.t[nt][v] - (dt / 3.0f) * k1p[c]);
          }
        }
    }
    St8 K3 = feval_staged(W1p, W2p, W3p, scr, b1r, b2r, b3r, l16, hi);

    // ---- stage s4 = y + dt*(k1 - k2 + k3); fold K2 <- k1 + 3*k2 + 3*k3. K3 dies.
    {
      const int zs = opaque_zero();
#pragma unroll
      for (int nt = 0; nt < 8; ++nt)
#pragma unroll
        for (int j = 0; j < 4; ++j) {
          const v2f yp  = ld2(yb, lane, nt, j, zs);
          const v2f k1p = ld2(k1b, lane, nt, j, zs);
#pragma unroll
          for (int c = 0; c < 2; ++c) {
            const int v = 2 * j + c;
            tstore(scrF, l16, hi, nt, v,
                   yp[c] + dt * (k1p[c] - K2.t[nt][v] + K3.t[nt][v]));
            K2.t[nt][v] = k1p[c] + 3.0f * K2.t[nt][v] + 3.0f * K3.t[nt][v];
          }
        }
    }
    St8 K4 = feval_staged(W1p, W2p, W3p, scr, b1r, b2r, b3r, l16, hi);

    // ---- y += dt/8*(k1+3k2+3k3+k4); park y, emit preds[...,step], stage next s1.
    {
      const int zs = opaque_zero();
#pragma unroll
      for (int nt = 0; nt < 8; ++nt)
#pragma unroll
        for (int j = 0; j < 4; ++j) {
          const v2f yp = ld2(yb, lane, nt, j, zs);
          float yn[2];
#pragma unroll
          for (int c = 0; c < 2; ++c) {
            const int v = 2 * j + c;
            yn[c] = yp[c] + (dt / 8.0f) * (K2.t[nt][v] + K4.t[nt][v]);
            out[(size_t)(m0 + v + 8 * hi) * (kN * kH) +
                (size_t)(nt * 16 + l16) * kH + step] = yn[c];
            tstore(scrF, l16, hi, nt, v, yn[c]);  // s1 for next step
          }
          st2(yb, lane, nt, j, yn[0], yn[1]);
        }
    }
  }
}

extern "C" void kernel_launch(void* const* d_in, const int* in_sizes, int n_in,
                              void* d_out, int out_size, void* d_ws, size_t ws_size,
                              hipStream_t stream) {
  (void)in_sizes; (void)n_in; (void)d_ws; (void)ws_size; (void)out_size;
  const float* x  = (const float*)d_in[0];
  const float* W1 = (const float*)d_in[1];
  const float* b1 = (const float*)d_in[2];
  const float* W2 = (const float*)d_in[3];
  const float* b2 = (const float*)d_in[4];
  const float* W3 = (const float*)d_in[5];
  const float* b3 = (const float*)d_in[6];
  float* out = (float*)d_out;

  // Allow >48KB dynamic LDS (gfx1250 has 320KB/WGP). Deterministic, capture-safe.
  (void)hipFuncSetAttribute(reinterpret_cast<const void*>(node_rk4_kernel),
                            hipFuncAttributeMaxDynamicSharedMemorySize, SMEM_BYTES);

  node_rk4_kernel<<<kGrid, kThreads, SMEM_BYTES, stream>>>(x, W1, b1, W2, b2, W3,
                                                           b3, out);
}